// QuantumAutoencoder_65481071402811
// MI455X (gfx1250) — compile-verified
//
#include <hip/hip_runtime.h>

typedef unsigned int u32;
typedef unsigned long long u64;
typedef __uint128_t u128;

// ---------------------------------------------------------------------------
// Compile-time reproduction of np.random.default_rng(1234) gate structure
// (SeedSequence pool mixing + PCG64 xsl-rr-128-64 + numpy's buffered 32-bit
// Lemire bounded draws + Generator.choice Floyd's algorithm w/ final shuffle)
// ---------------------------------------------------------------------------
namespace qrng {

constexpr u32 XSHIFT = 16;
constexpr u32 INIT_A = 0x43b0d7e5u;
constexpr u32 MULT_A = 0x931e8875u;
constexpr u32 INIT_B = 0x8b51f9ddu;
constexpr u32 MULT_B = 0x58f38dedu;
constexpr u32 MIX_L  = 0xca01f9ddu;
constexpr u32 MIX_R  = 0x4973f715u;

struct Pcg { u128 state; u128 inc; bool has32; u32 buf; };

constexpr u32 hashmix(u32 v, u32& hc) { v ^= hc; hc *= MULT_A; v *= hc; v ^= v >> XSHIFT; return v; }
constexpr u32 mixfn(u32 x, u32 y) { u32 r = (MIX_L * x) ^ (MIX_R * y); r ^= r >> XSHIFT; return r; }
constexpr u128 pcg_mul() { return ((u128)0x2360ed051fc65da4ULL << 64) | 0x4385df649fccf645ULL; }

constexpr Pcg make_pcg(u32 entropy0) {
  u32 pool[4] = {0, 0, 0, 0};
  u32 hc = INIT_A;
  pool[0] = hashmix(entropy0, hc);
  for (int i = 1; i < 4; ++i) pool[i] = hashmix(0u, hc);
  for (int s = 0; s < 4; ++s)
    for (int d = 0; d < 4; ++d)
      if (s != d) pool[d] = mixfn(pool[d], hashmix(pool[s], hc));
  // generate_state(4, uint64) -> 8 uint32 words viewed little-endian as 4 u64
  u32 w[8] = {};
  u32 hb = INIT_B;
  for (int i = 0; i < 8; ++i) {
    u32 dv = pool[i & 3];
    dv ^= hb; hb *= MULT_B; dv *= hb; dv ^= dv >> XSHIFT;
    w[i] = dv;
  }
  u64 v0 = (u64)w[0] | ((u64)w[1] << 32);
  u64 v1 = (u64)w[2] | ((u64)w[3] << 32);
  u64 v2 = (u64)w[4] | ((u64)w[5] << 32);
  u64 v3 = (u64)w[6] | ((u64)w[7] << 32);
  u128 initstate = ((u128)v0 << 64) | v1;   // seed[0] = high
  u128 initseq   = ((u128)v2 << 64) | v3;
  Pcg p{};
  p.state = 0;
  p.inc = (initseq << 1) | 1;
  p.state = p.state * pcg_mul() + p.inc;
  p.state += initstate;
  p.state = p.state * pcg_mul() + p.inc;
  p.has32 = false; p.buf = 0;
  return p;
}

constexpr u64 next64(Pcg& p) {
  p.state = p.state * pcg_mul() + p.inc;
  u32 rot = (u32)(p.state >> 122) & 63u;
  u64 x = (u64)(p.state >> 64) ^ (u64)p.state;
  return rot ? ((x >> rot) | (x << (64 - rot))) : x;
}
constexpr u32 next32(Pcg& p) {
  if (p.has32) { p.has32 = false; return p.buf; }
  u64 n = next64(p);
  p.has32 = true; p.buf = (u32)(n >> 32);
  return (u32)n;
}
// numpy buffered_bounded_lemire_uint32, rng = inclusive upper bound
constexpr u32 lemire(Pcg& p, u32 rng) {
  u32 re = rng + 1u;
  u64 m = (u64)next32(p) * (u64)re;
  u32 left = (u32)m;
  if (left < re) {
    u32 thr = (0xFFFFFFFFu - rng) % re;
    while (left < thr) { m = (u64)next32(p) * (u64)re; left = (u32)m; }
  }
  return (u32)(m >> 32);
}

struct Spec { int type[50]; int w0[50]; int w1[50]; };

constexpr Spec make_spec() {
  Pcg p = make_pcg(1234u);
  Spec s{};
  for (int i = 0; i < 50; ++i) s.type[i] = (int)lemire(p, 3u);   // integers(0,4,size=50)
  for (int i = 0; i < 50; ++i) {
    if (s.type[i] < 3) {
      s.w0[i] = (int)lemire(p, 13u);                             // integers(0,14)
      s.w1[i] = -1;
    } else {
      // choice(14, size=2, replace=False): Floyd's (hash set size 4, mask 3)
      u64 hs[4] = {(u64)-1, (u64)-1, (u64)-1, (u64)-1};
      u64 idx[2] = {0, 0};
      for (int j = 12; j < 14; ++j) {
        u64 val = (u64)lemire(p, (u32)j);
        u64 loc = val & 3u;
        while (hs[loc] != (u64)-1 && hs[loc] != val) loc = (loc + 1) & 3u;
        if (hs[loc] == (u64)-1) { hs[loc] = val; idx[j - 12] = val; }
        else {
          loc = (u64)j & 3u;
          while (hs[loc] != (u64)-1) loc = (loc + 1) & 3u;
          hs[loc] = (u64)j; idx[j - 12] = (u64)j;
        }
      }
      u32 jj = lemire(p, 1u);   // _shuffle_int(2,1): swap idx[jj] <-> idx[1]
      u64 t = idx[jj]; idx[jj] = idx[1]; idx[1] = t;
      s.w0[i] = (int)idx[0];    // control wire
      s.w1[i] = (int)idx[1];    // target wire
    }
  }
  return s;
}

} // namespace qrng

__constant__ qrng::Spec G_SPEC = qrng::make_spec();

// ---------------------------------------------------------------------------
// Kernel 1: LDS-resident state-vector simulation, one batch element per block
//   state: 16384 complex64 = 128 KB dynamic LDS (2 workgroups / 320KB WGP)
//   writes S[b][h] = sum_{l<64} |amp[h*64+l]|^2  (top-8-bit partial sums)
// ---------------------------------------------------------------------------
#define NW     14
#define NST    16384
#define NPAIR  8192

__global__ __launch_bounds__(256) void qae_sim(const float* __restrict__ x,
                                               const float* __restrict__ rl,
                                               float* __restrict__ S) {
  extern __shared__ float2 st[];
  const int tid = threadIdx.x;
  const int b = blockIdx.x;

  for (int i = tid; i < NST; i += 256) st[i] = make_float2(0.f, 0.f);
  if (tid == 0) st[0] = make_float2(1.f, 0.f);   // |0...0>
  __syncthreads();

  // ---- angle encoding: RX(x[b,i]) on wire i (bit 13-i) ----
  for (int wi = 0; wi < NW; ++wi) {
    float sn, cs;
    sincosf(0.5f * x[b * NW + wi], &sn, &cs);
    const int bit = 13 - wi;
    const int lo = (1 << bit) - 1;
    for (int p = tid; p < NPAIR; p += 256) {
      int i0 = ((p >> bit) << (bit + 1)) | (p & lo);
      int i1 = i0 | (1 << bit);
      float2 a0 = st[i0], a1 = st[i1];
      float2 r0, r1;
      r0.x = cs * a0.x + sn * a1.y;  r0.y = cs * a0.y - sn * a1.x;
      r1.x = cs * a1.x + sn * a0.y;  r1.y = cs * a1.y - sn * a0.x;
      st[i0] = r0; st[i1] = r1;
    }
    __syncthreads();
  }

  // ---- random layer: 50 fixed-structure gates ----
  for (int k = 0; k < 50; ++k) {
    const int t = G_SPEC.type[k];
    float sn, cs;
    sincosf(0.5f * rl[k], &sn, &cs);
    if (t == 3) {  // CRX(control=w0, target=w1)
      const int cbit = 13 - G_SPEC.w0[k];
      const int tbit = 13 - G_SPEC.w1[k];
      const int bl = cbit < tbit ? cbit : tbit;
      const int bh = cbit < tbit ? tbit : cbit;
      const int lol = (1 << bl) - 1, loh = (1 << bh) - 1;
      for (int p = tid; p < 4096; p += 256) {
        int i = ((p >> bl) << (bl + 1)) | (p & lol);
        i = ((i >> bh) << (bh + 1)) | (i & loh);
        int i0 = i | (1 << cbit);           // control bit = 1
        int i1 = i0 | (1 << tbit);
        float2 a0 = st[i0], a1 = st[i1];
        float2 r0, r1;
        r0.x = cs * a0.x + sn * a1.y;  r0.y = cs * a0.y - sn * a1.x;
        r1.x = cs * a1.x + sn * a0.y;  r1.y = cs * a1.y - sn * a0.x;
        st[i0] = r0; st[i1] = r1;
      }
    } else {
      const int bit = 13 - G_SPEC.w0[k];
      const int lo = (1 << bit) - 1;
      for (int p = tid; p < NPAIR; p += 256) {
        int i0 = ((p >> bit) << (bit + 1)) | (p & lo);
        int i1 = i0 | (1 << bit);
        float2 a0 = st[i0], a1 = st[i1];
        float2 r0, r1;
        if (t == 0) {            // RX
          r0.x = cs * a0.x + sn * a1.y;  r0.y = cs * a0.y - sn * a1.x;
          r1.x = cs * a1.x + sn * a0.y;  r1.y = cs * a1.y - sn * a0.x;
        } else if (t == 1) {     // RY
          r0.x = cs * a0.x - sn * a1.x;  r0.y = cs * a0.y - sn * a1.y;
          r1.x = sn * a0.x + cs * a1.x;  r1.y = sn * a0.y + cs * a1.y;
        } else {                 // RZ: diag(e^{-i t/2}, e^{+i t/2})
          r0.x = cs * a0.x + sn * a0.y;  r0.y = cs * a0.y - sn * a0.x;
          r1.x = cs * a1.x - sn * a1.y;  r1.y = cs * a1.y + sn * a1.x;
        }
        st[i0] = r0; st[i1] = r1;
      }
    }
    __syncthreads();
  }

  // ---- partial reduction over low 6 bits: h = tid in [0,256) ----
  float acc = 0.f;
  #pragma unroll 8
  for (int l = 0; l < 64; ++l) {
    float2 a = st[tid * 64 + l];
    acc = fmaf(a.x, a.x, fmaf(a.y, a.y, acc));
  }
  S[b * 256 + tid] = acc;
}

// ---------------------------------------------------------------------------
// Kernel 2: expz[256,8] = S[256,256] @ Zsign[256,16] on the matrix pipe
//   V_WMMA_F32_16X16X4_F32, one wave per 16-row M-tile, K=256 (64 WMMAs).
//   B (+/-1 signs for wires 0..7, zero pad for 8..15) generated branchlessly
//   in registers so the WMMA loop body has uniform EXEC (no saveexec diamonds)
//   and the A-tile global_load_b64 is software-pipelined one step ahead.
//   A layout: M=lane%16, K pair = (lane/16)*2 + {vgpr0,vgpr1}
//   B/D layout: N=lane%16, rows split across lane halves.
// ---------------------------------------------------------------------------
typedef float v2f __attribute__((ext_vector_type(2)));
typedef float v8f __attribute__((ext_vector_type(8)));

__global__ __launch_bounds__(32) void qae_expz(const float* __restrict__ S,
                                               float* __restrict__ out) {
  const int lane = threadIdx.x;
  const int tile = blockIdx.x;         // 16 tiles x 16 batch rows
  const int n = lane & 15;             // column (wire) for B/D, row M for A
  const int kh = (lane >> 4) * 2;      // K sub-offset within 4-wide chunk
  const int arow = tile * 16 + n;      // A row (M = lane%16)

  // Branchless per-lane column state (uniform EXEC inside the K loop):
  //   wires 0..7: sign = 1 - 2*bit(7-n of h);  pad columns 8..15: 0.0
  const float colmask = (n < 8) ? 1.f : 0.f;
  const int sh = 7 - (n & 7);          // clamped shift, masked by colmask

  v8f c = {0.f, 0.f, 0.f, 0.f, 0.f, 0.f, 0.f, 0.f};
  const float2* __restrict__ Sv = reinterpret_cast<const float2*>(S);
  const int sbase = (arow * 256 + kh) >> 1;   // float2 index at k0 = 0

  float2 av = Sv[sbase];                      // prefetch first A pair
  #pragma unroll 4
  for (int k0 = 0; k0 < 256; k0 += 4) {
    const int kA = k0 + kh;
    // sign columns, computed with pure VALU (no divergence)
    v2f bm;
    bm.x = colmask * (1.f - 2.f * (float)((kA >> sh) & 1));
    bm.y = colmask * (1.f - 2.f * (float)(((kA + 1) >> sh) & 1));
    v2f a; a.x = av.x; a.y = av.y;
    // prefetch next iteration's A pair (wraps harmlessly on last iter)
    const int knext = (k0 + 4) & 255;
    av = Sv[sbase + (knext >> 1)];
    c = __builtin_amdgcn_wmma_f32_16x16x4_f32(false, a, false, bm,
                                              (short)0, c, false, false);
  }

  // D: VGPR j -> M=j (lanes 0-15) / M=j+8 (lanes 16-31), N = lane&15
  if (n < 8) {
    const int mbase = (lane < 16) ? 0 : 8;
    #pragma unroll
    for (int j = 0; j < 8; ++j)
      out[(tile * 16 + mbase + j) * 8 + n] = c[j];
  }
}

// ---------------------------------------------------------------------------
extern "C" void kernel_launch(void* const* d_in, const int* in_sizes, int n_in,
                              void* d_out, int out_size, void* d_ws, size_t ws_size,
                              hipStream_t stream) {
  const float* x  = (const float*)d_in[0];   // [256,14] f32
  const float* rl = (const float*)d_in[1];   // [50] f32
  float* S   = (float*)d_ws;                 // [256,256] f32 partial sums (256 KB)
  float* out = (float*)d_out;                // [256,8] f32

  qae_sim<<<256, 256, NST * sizeof(float2), stream>>>(x, rl, S);
  qae_expz<<<16, 32, 0, stream>>>(S, out);
}